// Calc_Xi_And_LogLikelihood_71777493451127
// MI455X (gfx1250) — compile-verified
//
#include <hip/hip_runtime.h>

// ---------------------------------------------------------------------------
// xi = mu + (sigma+1e-5)*eps                        (64,16,200,64) f32
// LL[b,c] = sum_{t,d} (-0.5*z^2 - log s) - TS*(D/2)*ln(2pi)   (64,16) f32
//
// Memory-bound: ~210 MB traffic -> ~9us @ 23.3 TB/s floor. Strategy:
//   * 1 block per (b,c): 12800 contiguous floats per input, streamed as f4
//   * non-temporal b128 loads/stores (no reuse; keep the 192MB L2 clean)
//   * z = (xi-mu)/s == eps exactly (xi = fma(s,eps,mu)) -> no division
//   * log s via raw v_log_f32 (log2) * ln2 -> no libm fixup code
//   * reduction accumulated on the matrix pipe: v_wmma_f32_16x16x4_f32
//     with all-ones B folds 64 f32 terms/wave/op into a persistent C
//   * compile-time trip count (10) -> scalar loop, EXEC stays all-ones
//     for WMMA (ISA requirement), no per-lane exec masking
// ---------------------------------------------------------------------------

typedef float f4  __attribute__((ext_vector_type(4)));
typedef float v2f __attribute__((ext_vector_type(2)));
typedef float v8f __attribute__((ext_vector_type(8)));

constexpr int BS = 64, NC = 16, TS = 200, D = 64;
constexpr int PER_BC     = TS * D;             // 12800 floats per (b,c)
constexpr int VEC_PER_BC = PER_BC / 4;         // 3200 float4s per (b,c)
constexpr int THREADS    = 320;                // 10 wave32s
constexpr int ITERS      = VEC_PER_BC / THREADS;  // == 10, exact
constexpr int WAVES      = THREADS / 32;

// TS * (D/2) * ln(2*pi) = 200 * 32 * 1.8378770664093453
constexpr float LOG_NORM_TOTAL = 11762.413225019810f;
constexpr float LN2 = 0.69314718055994531f;

__global__ __launch_bounds__(THREADS)
void Calc_Xi_And_LogLikelihood_kernel(const float* __restrict__ mu,
                                      const float* __restrict__ sigma,
                                      const float* __restrict__ eps,
                                      float* __restrict__ xi,
                                      float* __restrict__ ll)
{
    const int bc = blockIdx.x;                       // 0..1023 = (b*NC + c)
    const long base = (long)bc * PER_BC;

    const f4* __restrict__ mu4 = (const f4*)(mu + base);
    const f4* __restrict__ sg4 = (const f4*)(sigma + base);
    const f4* __restrict__ ep4 = (const f4*)(eps + base);
    f4* __restrict__ xi4 = (f4*)(xi + base);

    const int tid = threadIdx.x;

    // Persistent WMMA accumulator: B = ones => D[m,n] = sum_k A[m,k] + C[m,n].
    v8f acc = {};
    v2f ones; ones[0] = 1.0f; ones[1] = 1.0f;

#pragma unroll 2
    for (int it = 0; it < ITERS; ++it) {
        const int i = tid + it * THREADS;

        f4 m = __builtin_nontemporal_load(mu4 + i);
        f4 s = __builtin_nontemporal_load(sg4 + i);
        f4 e = __builtin_nontemporal_load(ep4 + i);

        s = s + 1e-5f;            // sigma offset; s in (1e-5, ~2): no denorms
        f4 x = m + s * e;         // reparameterization (fma)
        __builtin_nontemporal_store(x, xi4 + i);

        // z = (xi - mu)/s = eps exactly; term = -0.5*e^2 - ln2*log2(s)
        f4 t;
#pragma unroll
        for (int k = 0; k < 4; ++k) {
            float lg = __builtin_amdgcn_logf(s[k]);            // v_log_f32
            t[k] = fmaf(-LN2, lg, -0.5f * (e[k] * e[k]));
        }

        // 4 terms/lane -> two 16x4 f32 A fragments (2 f32/lane each),
        // both chained into the same C on the matrix pipe.
        v2f a0 = {t[0], t[1]};
        v2f a1 = {t[2], t[3]};
        acc = __builtin_amdgcn_wmma_f32_16x16x4_f32(
                  false, a0, false, ones, (short)0, acc, false, false);
        acc = __builtin_amdgcn_wmma_f32_16x16x4_f32(
                  false, a1, false, ones, (short)0, acc, false, false);
    }

    // C holds every row-sum replicated across 16 columns: sum of all 256
    // entries = 16 * (true wave sum). Reduce 8 VGPRs/lane, then the wave.
    float local = 0.0f;
#pragma unroll
    for (int k = 0; k < 8; ++k) local += acc[k];

    float wsum = local;
#pragma unroll
    for (int off = 16; off > 0; off >>= 1)
        wsum += __shfl_down(wsum, off, 32);

    __shared__ float red[WAVES];
    const int wave = tid >> 5, lane = tid & 31;
    if (lane == 0) red[wave] = wsum * 0.0625f;   // /16, exact
    __syncthreads();

    if (tid == 0) {
        float tot = 0.0f;
#pragma unroll
        for (int w = 0; w < WAVES; ++w) tot += red[w];
        ll[bc] = tot - LOG_NORM_TOTAL;
    }
}

extern "C" void kernel_launch(void* const* d_in, const int* in_sizes, int n_in,
                              void* d_out, int out_size, void* d_ws, size_t ws_size,
                              hipStream_t stream) {
    const float* mu    = (const float*)d_in[0];
    const float* sigma = (const float*)d_in[1];
    const float* eps   = (const float*)d_in[2];

    float* xi = (float*)d_out;                         // 13,107,200 floats
    float* ll = xi + (size_t)BS * NC * TS * D;         // + 1,024 floats

    hipLaunchKernelGGL(Calc_Xi_And_LogLikelihood_kernel,
                       dim3(BS * NC), dim3(THREADS), 0, stream,
                       mu, sigma, eps, xi, ll);
}